// Extract_Overall_81819126988983
// MI455X (gfx1250) — compile-verified
//
#include <hip/hip_runtime.h>

// Weighted segment-mean over user neighborhoods, reduced to a single [128]
// vector.  row_ids == repeat(arange(U), 32) -> each user's 32 edges are
// contiguous.  Per-user normalization folds into per-edge weights, turning
// the whole problem into one weighted gather-reduce:
//     out[h] = (1/U) * sum_e w'_e * F[col_e, h]
// mapped onto V_WMMA_F32_16X16X4_F32 (row M=0 of A = 4 edge weights,
// B = 4 gathered feature rows x 16 hidden columns, f32 accumulate).
//
// Round-2 changes (latency, not bandwidth, is the per-wave bottleneck):
//  * all A operands + row offsets for a user are precomputed up front
//    (32 clause-able ds_bpermutes), removing shuffles from the hot loop
//  * B operands are double-buffered: group g+1's 16 loads are issued before
//    group g's 8 WMMAs, so only one L2 round trip is exposed per user
//  * 32-bit element offsets (table is 51 MB) -> GVS addressing, no 64-bit
//    shift/add pairs per row

#define HIDDEN 128
#define DEG 32
#define NGROUP (DEG / 4)          // 8 K=4 edge groups per user
#define NCHUNK (HIDDEN / 16)      // 8 accumulators of 16 columns each
#define BLOCK_THREADS 256
#define WAVES_PER_BLOCK (BLOCK_THREADS / 32)

typedef float v2f __attribute__((ext_vector_type(2)));
typedef float v8f __attribute__((ext_vector_type(8)));

__global__ __launch_bounds__(BLOCK_THREADS)
void seg_wmma_kernel(const float* __restrict__ feature,   // [N_ITEMS, 128]
                     const float* __restrict__ values,    // [U*32]
                     const int*   __restrict__ col_ids,   // [U*32]
                     float*       __restrict__ partials,  // [gridDim.x, 128]
                     int n_users)
{
    __shared__ float smem[WAVES_PER_BLOCK * HIDDEN];

    const int lane       = threadIdx.x & 31;
    const int waveInBlk  = threadIdx.x >> 5;
    const int gw         = blockIdx.x * WAVES_PER_BLOCK + waveInBlk;
    const int waveStride = gridDim.x * WAVES_PER_BLOCK;

    const int nsub = lane & 15;          // N-column within a 16-wide chunk
    const int ksel = (lane >> 4) << 1;   // lanes 0-15 -> K0/K1, lanes 16-31 -> K2/K3

    v8f acc[NCHUNK] = {};                // C/D accumulators (row 0 is live)

    for (int u = gw; u < n_users; u += waveStride) {
        const int e = u * DEG + lane;
        float v = values[e];             // coalesced 128B per wave
        int   c = col_ids[e];

        // wave32 sum of this user's 32 ratings
        float vs = v;
        #pragma unroll
        for (int off = 16; off > 0; off >>= 1)
            vs += __shfl_xor(vs, off, 32);
        const float w = v / vs;          // normalized edge weight

        // ---- precompute all A operands and B row offsets (clause of
        //      bpermutes, off the critical load->wmma path) ----
        v2f      Aops[NGROUP];
        unsigned o0[NGROUP], o1[NGROUP];   // float-element offsets (< 2^25)
        #pragma unroll
        for (int g = 0; g < NGROUP; ++g) {
            const int kbase = g * 4 + ksel;
            // A row M=0 only: lane 0 = w[k0],w[k1]; lane 16 = w[k2],w[k3]
            float a0 = __shfl(w, kbase,     32);
            float a1 = __shfl(w, kbase + 1, 32);
            if (nsub != 0) { a0 = 0.0f; a1 = 0.0f; }
            Aops[g].x = a0; Aops[g].y = a1;
            o0[g] = ((unsigned)__shfl(c, kbase,     32) << 7) + (unsigned)nsub;
            o1[g] = ((unsigned)__shfl(c, kbase + 1, 32) << 7) + (unsigned)nsub;
        }

        // ---- software-pipelined gather + WMMA: load group g+1 while
        //      multiplying group g ----
        v2f Bbuf[2][NCHUNK];
        #pragma unroll
        for (int j = 0; j < NCHUNK; ++j) {          // prologue: group 0
            Bbuf[0][j].x = feature[o0[0] + j * 16];
            Bbuf[0][j].y = feature[o1[0] + j * 16];
        }
        #pragma unroll
        for (int g = 0; g < NGROUP; ++g) {
            const int cur = g & 1, nxt = cur ^ 1;
            if (g + 1 < NGROUP) {
                #pragma unroll
                for (int j = 0; j < NCHUNK; ++j) {
                    Bbuf[nxt][j].x = feature[o0[g + 1] + j * 16];
                    Bbuf[nxt][j].y = feature[o1[g + 1] + j * 16];
                }
            }
            #pragma unroll
            for (int j = 0; j < NCHUNK; ++j) {
                acc[j] = __builtin_amdgcn_wmma_f32_16x16x4_f32(
                    /*neg_a=*/false, Aops[g], /*neg_b=*/false, Bbuf[cur][j],
                    /*c_mod=*/(short)0, acc[j],
                    /*reuse_a=*/false, /*reuse_b=*/false);
            }
        }
    }

    // Row M=0 of C lives in VGPR 0 on lanes 0..15 (N = lane).
    if (lane < 16) {
        #pragma unroll
        for (int j = 0; j < NCHUNK; ++j)
            smem[waveInBlk * HIDDEN + j * 16 + lane] = acc[j][0];
    }
    __syncthreads();

    // Deterministic fixed-order reduction across the block's waves.
    for (int h = threadIdx.x; h < HIDDEN; h += blockDim.x) {
        float s = 0.0f;
        #pragma unroll
        for (int wv = 0; wv < WAVES_PER_BLOCK; ++wv)
            s += smem[wv * HIDDEN + h];
        partials[(size_t)blockIdx.x * HIDDEN + h] = s;
    }
}

__global__ void finalize_kernel(const float* __restrict__ partials,
                                float* __restrict__ out,
                                int n_blocks, float inv_users)
{
    const int h = threadIdx.x;
    if (h < HIDDEN) {
        float s = 0.0f;
        for (int b = 0; b < n_blocks; ++b)      // fixed order: deterministic
            s += partials[(size_t)b * HIDDEN + h];
        out[h] = s * inv_users;
    }
}

extern "C" void kernel_launch(void* const* d_in, const int* in_sizes, int n_in,
                              void* d_out, int out_size, void* d_ws, size_t ws_size,
                              hipStream_t stream)
{
    const float* feature = (const float*)d_in[0];
    const float* values  = (const float*)d_in[1];
    // d_in[2] = row_ids: unused, edges are contiguous (DEG per user) by construction
    const int*   col_ids = (const int*)d_in[3];

    const int E       = in_sizes[1];
    const int n_users = E / DEG;

    int nBlocks = 512;                                    // 4096 waves for L2-gather latency hiding
    while ((size_t)nBlocks * HIDDEN * sizeof(float) > ws_size && nBlocks > 1)
        nBlocks >>= 1;

    float* partials = (float*)d_ws;                       // fully overwritten every call

    seg_wmma_kernel<<<nBlocks, BLOCK_THREADS, 0, stream>>>(
        feature, values, col_ids, partials, n_users);
    finalize_kernel<<<1, HIDDEN, 0, stream>>>(
        partials, (float*)d_out, nBlocks, 1.0f / (float)n_users);
}